// MarchingTetrahedraHelper_54906861912686
// MI455X (gfx1250) — compile-verified
//
#include <hip/hip_runtime.h>
#include <cmath>

#define TB 256

typedef unsigned int u32x4 __attribute__((ext_vector_type(4)));
typedef unsigned int u32x8 __attribute__((ext_vector_type(8)));

// Marching-tets tables (match reference exactly)
__constant__ int c_tri_table[16][6] = {
  {-1,-1,-1,-1,-1,-1},{1,0,2,-1,-1,-1},{4,0,3,-1,-1,-1},{1,4,2,1,3,4},
  {3,1,5,-1,-1,-1},{2,3,0,2,5,3},{1,4,0,1,5,4},{4,2,5,-1,-1,-1},
  {4,5,2,-1,-1,-1},{4,1,0,4,5,1},{3,2,0,3,5,2},{1,3,5,-1,-1,-1},
  {4,1,2,4,3,1},{3,0,4,-1,-1,-1},{2,0,1,-1,-1,-1},{-1,-1,-1,-1,-1,-1}};
__constant__ int c_num_tri[16] = {0,1,1,2,1,2,2,1,1,2,2,1,2,1,1,0};

// ---------------------------------------------------------------------------
// Canonical edge decode: eid = a*7 + (d-1), d in 1..7 encodes (dx,dy,dz) bits.
// Enumeration order equals lexicographic (min,max) order == jnp.unique order.
// ---------------------------------------------------------------------------
__device__ __forceinline__ int edge_decode(int eid, const unsigned char* occ,
                                           int N, int* aOut, int* bOut) {
  int a = eid / 7;
  int d = eid - a * 7 + 1;
  int NN = N * N;
  int x = a / NN;
  int r = a - x * NN;
  int y = r / N;
  int z = r - y * N;
  bool ok = true;
  if (d & 4) ok = ok && (x < N - 1);
  if (d & 2) ok = ok && (y < N - 1);
  if (d & 1) ok = ok && (z < N - 1);
  if (!ok) return 0;
  int b = a + ((d >> 2) & 1) * NN + ((d >> 1) & 1) * N + (d & 1);
  *aOut = a; *bOut = b;
  return (occ[a] != occ[b]) ? 1 : 0;
}

// ---------------------------------------------------------------------------
// Vectorized output init: verts region -> 0x0 (float 0.0f), rest -> -1.
// 164 MB stream => use b128 stores.
// ---------------------------------------------------------------------------
__global__ void k_init_out(int* out, long long nVertElems, long long nTotal) {
  long long nChunks = nTotal >> 2;
  long long stride = (long long)gridDim.x * blockDim.x;
  long long t0 = (long long)blockIdx.x * blockDim.x + threadIdx.x;
  for (long long c = t0; c < nChunks; c += stride) {
    long long i = c << 2;
    int4 val;
    if (i + 4 <= nVertElems) {
      val = make_int4(0, 0, 0, 0);
    } else if (i >= nVertElems) {
      val = make_int4(-1, -1, -1, -1);
    } else {
      val.x = (i + 0 < nVertElems) ? 0 : -1;
      val.y = (i + 1 < nVertElems) ? 0 : -1;
      val.z = (i + 2 < nVertElems) ? 0 : -1;
      val.w = (i + 3 < nVertElems) ? 0 : -1;
    }
    reinterpret_cast<int4*>(out)[c] = val;
  }
  long long ti = (nChunks << 2) + t0;
  if (ti < nTotal) out[ti] = (ti < nVertElems) ? 0 : -1;
}

__global__ void k_pos_occ(const float* __restrict__ level,
                          const float* __restrict__ def,
                          const float* __restrict__ gv,
                          float* __restrict__ pos,
                          unsigned char* __restrict__ occ,
                          float inv_res, int NV) {
  int v = blockIdx.x * TB + threadIdx.x;
  if (v >= NV) return;
  float s = level[v];
  occ[v] = (s > 0.0f) ? 1 : 0;
  #pragma unroll
  for (int c = 0; c < 3; ++c)
    pos[3 * v + c] = gv[3 * v + c] + inv_res * tanhf(def[3 * v + c]);
}

__global__ void k_edge_reduce(const unsigned char* __restrict__ occ,
                              const float* __restrict__ pos,
                              int* __restrict__ edgeSums, int N, int NE) {
  __shared__ int sh[TB];
  int tid = threadIdx.x;
  int eid = blockIdx.x * TB + tid;
  int flag = 0, a = 0, b = 0;
  if (eid < NE) {
    flag = edge_decode(eid, occ, N, &a, &b);
    if (flag) __builtin_prefetch(pos + 3 * (size_t)a, 0, 0);  // warm 192MB L2 for emit
  }
  sh[tid] = flag; __syncthreads();
  for (int s = TB / 2; s > 0; s >>= 1) {
    if (tid < s) sh[tid] += sh[tid + s];
    __syncthreads();
  }
  if (tid == 0) edgeSums[blockIdx.x] = sh[0];
}

__global__ void k_scan_i32(int* sums, int nb, int* countSlot) {
  __shared__ int sh[TB];
  int tid = threadIdx.x;
  int running = 0;
  for (int base = 0; base < nb; base += TB) {
    int i = base + tid;
    int v = (i < nb) ? sums[i] : 0;
    sh[tid] = v; __syncthreads();
    for (int off = 1; off < TB; off <<= 1) {
      int t = (tid >= off) ? sh[tid - off] : 0;
      __syncthreads();
      sh[tid] += t;
      __syncthreads();
    }
    if (i < nb) sums[i] = running + (sh[tid] - v);
    int chunkTotal = sh[TB - 1];
    __syncthreads();
    running += chunkTotal;
  }
  if (tid == 0) *countSlot = running;                // n_verts
}

__global__ void k_edge_emit(const unsigned char* __restrict__ occ,
                            const float* __restrict__ level,
                            const float* __restrict__ pos,
                            const int* __restrict__ edgeSums,
                            int* __restrict__ idxMap,
                            float* __restrict__ vertsOut, int N, int NE) {
  __shared__ int sh[TB];
  int tid = threadIdx.x;
  int eid = blockIdx.x * TB + tid;
  int flag = 0, a = 0, b = 0;
  if (eid < NE) flag = edge_decode(eid, occ, N, &a, &b);
  sh[tid] = flag; __syncthreads();
  for (int off = 1; off < TB; off <<= 1) {
    int t = (tid >= off) ? sh[tid - off] : 0;
    __syncthreads();
    sh[tid] += t;
    __syncthreads();
  }
  int excl = sh[tid] - flag;
  if (eid >= NE) return;
  if (flag) {
    int row = edgeSums[blockIdx.x] + excl;
    float s0 = level[a], s1 = level[b];
    float dnm = s0 - s1;
    float safe = (dnm == 0.0f) ? 1.0f : dnm;
    float w0 = -s1 / safe, w1 = s0 / safe;
    #pragma unroll
    for (int c = 0; c < 3; ++c)
      vertsOut[3 * (size_t)row + c] = pos[3 * a + c] * w0 + pos[3 * b + c] * w1;
    idxMap[eid] = row;
  } else {
    idxMap[eid] = -1;
  }
}

// ---------------------------------------------------------------------------
// Tet reduce: per-lane CDNA5 async global->LDS DMA staging (ASYNCcnt path).
// Single __shared__ object => group-segment offset 0 => LDS dest = tid*16.
// ---------------------------------------------------------------------------
__global__ void k_tet_reduce(const int* __restrict__ tets,
                             const unsigned char* __restrict__ occ,
                             unsigned long long* __restrict__ tetSums, int F) {
  __shared__ long long smem[512 + 256];            // [0,4KB): staging; then u64 reduce
  int* stag = reinterpret_cast<int*>(smem);
  unsigned long long* red = reinterpret_cast<unsigned long long*>(smem + 512);
  int tid = threadIdx.x;
  int tet = blockIdx.x * TB + tid;
  if (tet < F) {
    unsigned ldsOff = (unsigned)tid * 16u;
    unsigned gOff   = (unsigned)tet * 16u;
    asm volatile("global_load_async_to_lds_b128 %0, %1, %2"
                 :: "v"(ldsOff), "v"(gOff), "s"(tets) : "memory");
  }
  asm volatile("s_wait_asynccnt 0x0" ::: "memory"); // each wave waits for its own DMA
  unsigned long long v = 0;
  if (tet < F) {
    int i0 = stag[tid * 4 + 0], i1 = stag[tid * 4 + 1];
    int i2 = stag[tid * 4 + 2], i3 = stag[tid * 4 + 3];
    int code = occ[i0] | (occ[i1] << 1) | (occ[i2] << 2) | (occ[i3] << 3);
    int nt = c_num_tri[code];
    v = (nt == 1) ? 1ULL : ((nt == 2) ? (1ULL << 32) : 0ULL);
  }
  red[tid] = v; __syncthreads();
  for (int s = TB / 2; s > 0; s >>= 1) {
    if (tid < s) red[tid] += red[tid + s];
    __syncthreads();
  }
  if (tid == 0) tetSums[blockIdx.x] = red[0];
}

__global__ void k_scan_u64(unsigned long long* sums, int nb,
                           int* nFacesSlot, int* nOneWs) {
  __shared__ unsigned long long sh[TB];
  int tid = threadIdx.x;
  unsigned long long running = 0;
  for (int base = 0; base < nb; base += TB) {
    int i = base + tid;
    unsigned long long v = (i < nb) ? sums[i] : 0ULL;
    sh[tid] = v; __syncthreads();
    for (int off = 1; off < TB; off <<= 1) {
      unsigned long long t = (tid >= off) ? sh[tid - off] : 0ULL;
      __syncthreads();
      sh[tid] += t;
      __syncthreads();
    }
    if (i < nb) sums[i] = running + (sh[tid] - v);
    unsigned long long chunkTotal = sh[TB - 1];
    __syncthreads();
    running += chunkTotal;
  }
  if (tid == 0) {
    unsigned nOne = (unsigned)(running & 0xffffffffULL);
    unsigned nTwo = (unsigned)(running >> 32);
    *nFacesSlot = (int)(nOne + 2u * nTwo);          // n_faces
    *nOneWs = (int)nOne;
  }
}

// ---------------------------------------------------------------------------
// Tet emit: stage the whole 4KB tile of tet indices with ONE Tensor Data Mover
// descriptor (TENSORcnt path). D# group0/group1 built in SGPRs; wave 0 issues,
// waits tensorcnt, then a workgroup barrier publishes LDS to all waves.
//   group0: count=1 | lds_addr=0 | global_addr=tile start | type=2
//   group1: data_size=4B, tile = 1024 x 1 elements, tensor_dim0 = remaining
//           elements (OOB reads return zero => generic partial tiles safe)
// ---------------------------------------------------------------------------
__global__ void k_tet_emit(const int* __restrict__ tets,
                           const unsigned char* __restrict__ occ,
                           const int* __restrict__ idxMap,
                           const unsigned long long* __restrict__ tetSums,
                           const int* __restrict__ nOneWs,
                           int* __restrict__ facesOut,
                           int* __restrict__ f2tOut, int N, int F) {
  __shared__ long long smem[512 + 256];
  int* stag = reinterpret_cast<int*>(smem);
  unsigned long long* sc = reinterpret_cast<unsigned long long*>(smem + 512);
  int tid = threadIdx.x;
  int tet = blockIdx.x * TB + tid;

  if (tid == 0) {
    unsigned long long ga =
        (unsigned long long)(const void*)tets + (unsigned long long)blockIdx.x * 4096ull;
    unsigned remain = (unsigned)F * 4u - (unsigned)blockIdx.x * 1024u;  // elements
    u32x4 g0;
    g0.x = 1u;                                          // count=1
    g0.y = 0u;                                          // lds_addr = 0 (staging base)
    g0.z = (unsigned)(ga & 0xffffffffull);              // global_addr[31:0]
    g0.w = (unsigned)((ga >> 32) & 0x1ffffffull) | (2u << 30);  // ga[56:32] | type=2
    u32x8 g1;
    g1.s0 = (2u << 16);                                 // data_size = 4 bytes
    g1.s1 = (remain & 0xffffu) << 16;                   // tensor_dim0[15:0]
    g1.s2 = (remain >> 16) | (1u << 16);                // tensor_dim0[31:16] | tensor_dim1=1
    g1.s3 = (1024u << 16);                              // tile_dim0 = 1024
    g1.s4 = 1u;                                         // tile_dim1 = 1, tile_dim2 = 0
    g1.s5 = remain;                                     // tensor_dim0_stride[31:0]
    g1.s6 = 0u;                                         // stride hi / dim1_stride lo
    g1.s7 = 0u;
    asm volatile("tensor_load_to_lds %0, %1" :: "s"(g0), "s"(g1) : "memory");
    __builtin_amdgcn_s_wait_tensorcnt(0);               // TDM complete for this wave
  }
  __syncthreads();                                      // publish LDS to all waves

  int ids[4] = {0, 0, 0, 0};
  int code = 0, nt = 0;
  unsigned long long v = 0;
  if (tet < F) {
    ids[0] = stag[tid * 4 + 0]; ids[1] = stag[tid * 4 + 1];
    ids[2] = stag[tid * 4 + 2]; ids[3] = stag[tid * 4 + 3];
    code = occ[ids[0]] | (occ[ids[1]] << 1) | (occ[ids[2]] << 2) | (occ[ids[3]] << 3);
    nt = c_num_tri[code];
    v = (nt == 1) ? 1ULL : ((nt == 2) ? (1ULL << 32) : 0ULL);
  }
  sc[tid] = v; __syncthreads();
  for (int off = 1; off < TB; off <<= 1) {
    unsigned long long t = (tid >= off) ? sc[tid - off] : 0ULL;
    __syncthreads();
    sc[tid] += t;
    __syncthreads();
  }
  unsigned long long pref = tetSums[blockIdx.x] + (sc[tid] - v);
  if (tet >= F || nt == 0) return;

  int NN = N * N;
  const int e0[6] = {0, 0, 0, 1, 1, 2};
  const int e1[6] = {1, 2, 3, 2, 3, 3};
  int em[6];
  #pragma unroll
  for (int j = 0; j < 6; ++j) {
    int a = ids[e0[j]], b = ids[e1[j]];
    if (a > b) { int t = a; a = b; b = t; }
    int diff = b - a;                         // in {1,N,N+1,NN,NN+1,NN+N,NN+N+1}
    int xb = (diff >= NN) ? 1 : 0;
    int rem = diff - xb * NN;
    int yb = (rem >= N) ? 1 : 0;
    rem -= yb * N;                            // rem in {0,1}
    int d = 4 * xb + 2 * yb + rem;
    em[j] = idxMap[a * 7 + d - 1];
  }
  int fv[6];
  #pragma unroll
  for (int j = 0; j < 6; ++j) {
    int t = c_tri_table[code][j];
    if (t < 0) t = 0;                         // jnp.clip(tt,0,5)
    fv[j] = em[t];
  }
  int n_one = *nOneWs;
  int r1 = (int)(pref & 0xffffffffULL);
  int r2 = (int)(pref >> 32);
  int p0 = (nt == 1) ? r1 : (n_one + 2 * r2);
  facesOut[3 * (size_t)p0 + 0] = fv[0];
  facesOut[3 * (size_t)p0 + 1] = fv[1];
  facesOut[3 * (size_t)p0 + 2] = fv[2];
  f2tOut[p0] = tet;
  if (nt == 2) {
    int p1 = n_one + 2 * r2 + 1;
    facesOut[3 * (size_t)p1 + 0] = fv[3];
    facesOut[3 * (size_t)p1 + 1] = fv[4];
    facesOut[3 * (size_t)p1 + 2] = fv[5];
    f2tOut[p1] = tet;
  }
}

// ---------------------------------------------------------------------------
// Host launcher
// ---------------------------------------------------------------------------
extern "C" void kernel_launch(void* const* d_in, const int* in_sizes, int n_in,
                              void* d_out, int out_size, void* d_ws, size_t ws_size,
                              hipStream_t stream) {
  (void)n_in; (void)out_size; (void)ws_size;
  const float* level = (const float*)d_in[0];
  const float* def   = (const float*)d_in[1];
  const float* gv    = (const float*)d_in[2];
  const int*   tets  = (const int*)d_in[3];

  int NV = in_sizes[0];
  int F  = in_sizes[3] / 4;
  int N  = (int)lround(cbrt((double)NV));     // grid side (res+1)
  float inv_res = 1.0f / (float)(N - 1);
  int NE = NV * 7;                            // canonical edge slots

  long long E = 6LL * F;                      // verts rows in reference output
  long long Fmax = 2LL * F;
  float* vertsOut = (float*)d_out;
  int* facesOut = (int*)d_out + 3 * E;
  int* f2tOut   = facesOut + 3 * Fmax;
  int* nvSlot   = f2tOut + Fmax;
  int* nfSlot   = nvSlot + 1;
  long long totalElems = 3 * E + 3 * Fmax + Fmax + 2;

  // Workspace carve (~12 MB)
  char* w = (char*)d_ws;
  size_t o = 0;
  auto A = [](size_t x) { return (x + 15) & ~(size_t)15; };
  float* pos = (float*)(w + o);                 o = A(o + (size_t)NV * 12);
  unsigned char* occ = (unsigned char*)(w + o); o = A(o + (size_t)NV);
  int* idxMap = (int*)(w + o);                  o = A(o + (size_t)NE * 4);
  int nbE = (NE + TB - 1) / TB;
  int nbT = (F + TB - 1) / TB;
  int* edgeSums = (int*)(w + o);                o = A(o + (size_t)nbE * 4);
  unsigned long long* tetSums = (unsigned long long*)(w + o); o = A(o + (size_t)nbT * 8);
  int* nOneWs = (int*)(w + o);                  o = A(o + 16);

  k_init_out<<<8192, TB, 0, stream>>>((int*)d_out, 3 * E, totalElems);
  k_pos_occ<<<(NV + TB - 1) / TB, TB, 0, stream>>>(level, def, gv, pos, occ, inv_res, NV);
  k_edge_reduce<<<nbE, TB, 0, stream>>>(occ, pos, edgeSums, N, NE);
  k_scan_i32<<<1, TB, 0, stream>>>(edgeSums, nbE, nvSlot);
  k_edge_emit<<<nbE, TB, 0, stream>>>(occ, level, pos, edgeSums, idxMap, vertsOut, N, NE);
  k_tet_reduce<<<nbT, TB, 0, stream>>>(tets, occ, tetSums, F);
  k_scan_u64<<<1, TB, 0, stream>>>(tetSums, nbT, nfSlot, nOneWs);
  k_tet_emit<<<nbT, TB, 0, stream>>>(tets, occ, idxMap, tetSums, nOneWs, facesOut, f2tOut, N, F);
}